// LidarFilter_58557584114405
// MI455X (gfx1250) — compile-verified
//
#include <hip/hip_runtime.h>
#include <hip/hip_bf16.h>
#include <math.h>

typedef __attribute__((ext_vector_type(2))) float v2f;
typedef __attribute__((ext_vector_type(8))) float v8f;

#define H2_   450
#define W_    1242
#define H_    375
#define NPTS  4000000
#define NPIX  (H2_ * W_)

// d_out layout (flat concat of reference tuple):
//   img:   [0, NPIX*7)
//   noocc: [NPIX*7, NPIX*7 + NPIX)
//   org:   [NPIX*8, NPIX*8 + NPIX)
//   filterIndicator (int32 bit pattern): [NPIX*9, NPIX*9 + NPTS)
#define OFF_IMG   0
#define OFF_NOOCC ((size_t)NPIX * 7)
#define OFF_ORG   ((size_t)NPIX * 8)
#define OFF_FILT  ((size_t)NPIX * 9)

// ---------------- setup: compute P_cam, P_lid, M = P_cam @ inv(P_lid) ----------
__device__ static void mat4_mul(const float* A, const float* B, float* C) {
    for (int i = 0; i < 4; ++i)
        for (int j = 0; j < 4; ++j) {
            float s = 0.f;
            for (int k = 0; k < 4; ++k) s += A[i * 4 + k] * B[k * 4 + j];
            C[i * 4 + j] = s;
        }
}

// Adjugate-based 4x4 inverse; valid for row-major input (inv(M^T)=(inv M)^T).
__device__ static void mat4_inv(const float* m, float* invOut) {
    float inv[16];
    inv[0]  =  m[5]*m[10]*m[15] - m[5]*m[11]*m[14] - m[9]*m[6]*m[15] + m[9]*m[7]*m[14] + m[13]*m[6]*m[11] - m[13]*m[7]*m[10];
    inv[4]  = -m[4]*m[10]*m[15] + m[4]*m[11]*m[14] + m[8]*m[6]*m[15] - m[8]*m[7]*m[14] - m[12]*m[6]*m[11] + m[12]*m[7]*m[10];
    inv[8]  =  m[4]*m[9]*m[15]  - m[4]*m[11]*m[13] - m[8]*m[5]*m[15] + m[8]*m[7]*m[13] + m[12]*m[5]*m[11] - m[12]*m[7]*m[9];
    inv[12] = -m[4]*m[9]*m[14]  + m[4]*m[10]*m[13] + m[8]*m[5]*m[14] - m[8]*m[6]*m[13] - m[12]*m[5]*m[10] + m[12]*m[6]*m[9];
    inv[1]  = -m[1]*m[10]*m[15] + m[1]*m[11]*m[14] + m[9]*m[2]*m[15] - m[9]*m[3]*m[14] - m[13]*m[2]*m[11] + m[13]*m[3]*m[10];
    inv[5]  =  m[0]*m[10]*m[15] - m[0]*m[11]*m[14] - m[8]*m[2]*m[15] + m[8]*m[3]*m[14] + m[12]*m[2]*m[11] - m[12]*m[3]*m[10];
    inv[9]  = -m[0]*m[9]*m[15]  + m[0]*m[11]*m[13] + m[8]*m[1]*m[15] - m[8]*m[3]*m[13] - m[12]*m[1]*m[11] + m[12]*m[3]*m[9];
    inv[13] =  m[0]*m[9]*m[14]  - m[0]*m[10]*m[13] - m[8]*m[1]*m[14] + m[8]*m[2]*m[13] + m[12]*m[1]*m[10] - m[12]*m[2]*m[9];
    inv[2]  =  m[1]*m[6]*m[15]  - m[1]*m[7]*m[14]  - m[5]*m[2]*m[15] + m[5]*m[3]*m[14] + m[13]*m[2]*m[7]  - m[13]*m[3]*m[6];
    inv[6]  = -m[0]*m[6]*m[15]  + m[0]*m[7]*m[14]  + m[4]*m[2]*m[15] - m[4]*m[3]*m[14] - m[12]*m[2]*m[7]  + m[12]*m[3]*m[6];
    inv[10] =  m[0]*m[5]*m[15]  - m[0]*m[7]*m[13]  - m[4]*m[1]*m[15] + m[4]*m[3]*m[13] + m[12]*m[1]*m[7]  - m[12]*m[3]*m[5];
    inv[14] = -m[0]*m[5]*m[14]  + m[0]*m[6]*m[13]  + m[4]*m[1]*m[14] - m[4]*m[2]*m[13] - m[12]*m[1]*m[6]  + m[12]*m[2]*m[5];
    inv[3]  = -m[1]*m[6]*m[11]  + m[1]*m[7]*m[10]  + m[5]*m[2]*m[11] - m[5]*m[3]*m[10] - m[9]*m[2]*m[7]   + m[9]*m[3]*m[6];
    inv[7]  =  m[0]*m[6]*m[11]  - m[0]*m[7]*m[10]  - m[4]*m[2]*m[11] + m[4]*m[3]*m[10] + m[8]*m[2]*m[7]   - m[8]*m[3]*m[6];
    inv[11] = -m[0]*m[5]*m[11]  + m[0]*m[7]*m[9]   + m[4]*m[1]*m[11] - m[4]*m[3]*m[9]  - m[8]*m[1]*m[7]   + m[8]*m[3]*m[5];
    inv[15] =  m[0]*m[5]*m[10]  - m[0]*m[6]*m[9]   - m[4]*m[1]*m[10] + m[4]*m[2]*m[9]  + m[8]*m[1]*m[6]   - m[8]*m[2]*m[5];
    float det = m[0]*inv[0] + m[1]*inv[4] + m[2]*inv[8] + m[3]*inv[12];
    float rdet = (det != 0.f) ? 1.f / det : 0.f;
    for (int i = 0; i < 16; ++i) invOut[i] = inv[i] * rdet;
}

// ws layout: [0..15] P_cam, [16..31] P_lid, [32..47] M
__global__ void setup_mats_kernel(const float* __restrict__ K,
                                  const float* __restrict__ camE,
                                  const float* __restrict__ lidE,
                                  float* __restrict__ mats) {
    if (threadIdx.x != 0 || blockIdx.x != 0) return;
    float Pcam[16], Plid[16], iPlid[16], M[16];
    mat4_mul(K, camE, Pcam);
    mat4_mul(K, lidE, Plid);
    mat4_inv(Plid, iPlid);
    mat4_mul(Pcam, iPlid, M);
    for (int i = 0; i < 16; ++i) {
        mats[i]      = Pcam[i];
        mats[16 + i] = Plid[i];
        mats[32 + i] = M[i];
    }
}

// ---------------- pixel init: back-project interp_depth through M --------------
__global__ __launch_bounds__(256) void pixel_init_kernel(
        const float* __restrict__ depth, const float* __restrict__ mats,
        float* __restrict__ img, float* __restrict__ noocc, float* __restrict__ orgm) {
    int i = blockIdx.x * blockDim.x + threadIdx.x;
    if (i >= NPIX) return;
    int row = i / W_;
    int col = i - row * W_;
    float d = depth[i];
    const float* M = mats + 32;
    float X = (float)col * d;
    float Y = (float)row * d;
    // pp = M @ [X, Y, d, 1]
    float pp0 = M[0] * X + M[1] * Y + M[2]  * d + M[3];
    float pp1 = M[4] * X + M[5] * Y + M[6]  * d + M[7];
    float pp2 = M[8] * X + M[9] * Y + M[10] * d + M[11];
    float ppz_safe = (pp2 != 0.f) ? pp2 : 1.f;
    float ppx = pp0 / ppz_safe;
    float ppy = pp1 / ppz_safe;
    bool dv = d > 0.f;
    float* o = img + (size_t)i * 7;
    o[0] = dv ? (float)col : 0.f;
    o[1] = dv ? (float)row : 0.f;
    o[2] = dv ? d          : 0.f;
    o[3] = dv ? ppx        : 0.f;
    o[4] = dv ? ppy        : 0.f;
    o[5] = dv ? pp2        : 0.f;
    o[6] = 0.f;
    noocc[i] = 0.f;
    orgm[i]  = 0.f;
}

// ---------------- zero filterIndicator ----------------------------------------
__global__ __launch_bounds__(256) void zero_filt_kernel(int* __restrict__ filt) {
    int i = blockIdx.x * blockDim.x + threadIdx.x;
    if (i < NPTS) filt[i] = 0;
}

// ---------------- point pass: WMMA batched 4x4 transform + scatter -------------
// A (16x4, rows 0-3 = P_cam, rows 4-7 = P_lid, rows 8-15 = 0) is constant.
// B (4x16) column n = [x,y,z,1] of point base+n.
// D column n = 8 transformed components of point n -> lane n (n<16) holds
// vp = P_cam*pt in c[0..2] and nv = P_lid*pt in c[4..6] -- no shuffles needed.
__global__ __launch_bounds__(256) void point_pass_kernel(
        const float* __restrict__ velo, const float* __restrict__ mats,
        float* __restrict__ img, float* __restrict__ noocc, float* __restrict__ orgm) {
    const int lane = threadIdx.x & 31;
    const int wave = (blockIdx.x * blockDim.x + threadIdx.x) >> 5;
    const int base = wave * 16;

    const int m    = lane & 15;        // row (A) / column (B,D) index
    const int koff = (lane >> 4) << 1; // 0 for lanes 0-15, 2 for lanes 16-31

    // A operand: per ISA F32 16x4 A layout, lane holds A[m][koff], A[m][koff+1]
    v2f a;
    a[0] = 0.f; a[1] = 0.f;
    if (m < 8) {
        const float* row = (m < 4) ? (mats + m * 4) : (mats + 16 + (m - 4) * 4);
        a[0] = row[koff];
        a[1] = row[koff + 1];
    }

    // B operand: lane supplies components koff, koff+1 of point base+m.
    // velo[:,3] == 1.0 supplies the homogeneous w row for free.
    const int p = base + m;
    const float* vptr = velo + (size_t)p * 4 + koff;
    v2f b;
    b[0] = vptr[0];
    b[1] = vptr[1];

    v8f c = {};
    c = __builtin_amdgcn_wmma_f32_16x16x4_f32(
            /*neg_a=*/false, a, /*neg_b=*/false, b,
            /*c_mod=*/(short)0, c, /*reuse_a=*/false, /*reuse_b=*/false);

    if (lane < 16) {
        float vpx = c[0], vpy = c[1], vpz = c[2];
        float nvx = c[4], nvy = c[5], nvz = c[6];
        float x0  = b[0];               // velo[p][0] (koff==0 on these lanes)

        float vx = vpx / vpz;
        float vy = vpy / vpz;
        bool onimg = (x0 > 0.f) && (vx > 0.f) && (vx < (float)W_) &&
                     (vy > 0.f) && (vy < (float)H_);

        float nx = nvx / nvz;
        float ny = nvy / nvz;
        float rnxf = rintf(nx);         // round-half-even, matches jnp.round
        float rnyf = rintf(ny);
        int rnx = (int)rnxf;
        int rny = (int)rnyf;
        bool nval = (rnx >= 0) && (rnx < W_) && (rny >= 0) && (rny < H2_) &&
                    (nvz > 0.f) && onimg;

        if (nval) {
            size_t pix = (size_t)rny * W_ + rnx;
            float* o = img + pix * 7;
            o[0] = nx;  o[1] = ny;  o[2] = nvz;
            o[3] = vx;  o[4] = vy;  o[5] = vpz;
            o[6] = (float)p;        // p < 2^24, exact in f32
            noocc[pix] = 1.f;
            orgm[pix]  = 1.f;
        }
    }
}

// ---------------- filter pass: gather scattered indices ------------------------
__global__ __launch_bounds__(256) void filter_pass_kernel(
        const float* __restrict__ img, const float* __restrict__ noocc,
        int* __restrict__ filt) {
    int i = blockIdx.x * blockDim.x + threadIdx.x;
    if (i >= NPIX) return;
    if (noocc[i] == 1.f) {
        int idx = (int)img[(size_t)i * 7 + 6];
        if (idx >= 0 && idx < NPTS) filt[idx] = 1;
    }
}

extern "C" void kernel_launch(void* const* d_in, const int* in_sizes, int n_in,
                              void* d_out, int out_size, void* d_ws, size_t ws_size,
                              hipStream_t stream) {
    const float* velo  = (const float*)d_in[0];
    const float* K     = (const float*)d_in[1];
    const float* camE  = (const float*)d_in[2];
    const float* lidE  = (const float*)d_in[3];
    const float* depth = (const float*)d_in[4];
    (void)in_sizes; (void)n_in; (void)out_size; (void)ws_size;

    float* out   = (float*)d_out;
    float* img   = out + OFF_IMG;
    float* noocc = out + OFF_NOOCC;
    float* orgm  = out + OFF_ORG;
    int*   filt  = (int*)(out + OFF_FILT); // int32 bit pattern, matches ref dtype

    float* mats = (float*)d_ws; // 48 floats

    setup_mats_kernel<<<1, 1, 0, stream>>>(K, camE, lidE, mats);

    const int TB = 256;
    pixel_init_kernel<<<(NPIX + TB - 1) / TB, TB, 0, stream>>>(depth, mats, img, noocc, orgm);
    zero_filt_kernel<<<(NPTS + TB - 1) / TB, TB, 0, stream>>>(filt);

    // 4M points, 16 per wave, 8 waves per 256-thread block -> 128 pts/block
    const int nblocks = NPTS / 128; // 31250, exact
    point_pass_kernel<<<nblocks, TB, 0, stream>>>(velo, mats, img, noocc, orgm);

    filter_pass_kernel<<<(NPIX + TB - 1) / TB, TB, 0, stream>>>(img, noocc, filt);
}